// Vanilla_68350109548796
// MI455X (gfx1250) — compile-verified
//
#include <hip/hip_runtime.h>

#define NNODES 50000
#define NEDGES 800000
#define HID 64
#define OUTD 32
#define NGRAPHS 256
#define NHOPS 3

typedef __attribute__((ext_vector_type(2))) float v2f;
typedef __attribute__((ext_vector_type(8))) float v8f;

// ---------------- degree / normalization ----------------

__global__ void deg_init_kernel(float* deg, int n) {
    int i = blockIdx.x * blockDim.x + threadIdx.x;
    if (i < n) deg[i] = 1.0f;  // self loop contribution
}

__global__ void deg_edges_kernel(const int* __restrict__ dst, float* deg, int E) {
    int e = blockIdx.x * blockDim.x + threadIdx.x;
    if (e < E) atomicAdd(&deg[dst[e]], 1.0f);
}

__global__ void dinv_kernel(float* deg, int n) {
    int i = blockIdx.x * blockDim.x + threadIdx.x;
    if (i < n) deg[i] = rsqrtf(deg[i]);  // deg >= 1 always (self loops)
}

// ---------------- WMMA fp32 GEMM: C[M x N] = op(A)[M x 64] * W[64 x N] (+ bias) ----------------
// Full fp32 via v_wmma_f32_16x16x4_f32. W staged in LDS once; each wave caches its 16
// B fragments in registers and reuses them across TILES_PER_WAVE row tiles. A rows are
// pointer-clamped at the M edge (garbage rows only affect garbage C rows, which are
// store-masked), so the K loop is completely branch-free: 16 batched global_load_b64
// then 16 back-to-back WMMAs.

template <int N, bool RELU, bool HASBIAS>
__global__ __launch_bounds__(256) void gemm_bias_kernel(
    const float* __restrict__ A, const float* __restrict__ W,
    const float* __restrict__ bias, float* __restrict__ C, int M) {
    constexpr int K = 64;
    constexpr int KSTEPS = K / 4;                 // 16
    constexpr int COL_TILES = N / 16;             // 4 (N=64) or 2 (N=32)
    constexpr int ROW_GROUPS = 8 / COL_TILES;     // 2 or 4
    constexpr int TILES_PER_WAVE = 2;
    constexpr int ROWS = 16 * ROW_GROUPS * TILES_PER_WAVE;  // 64 or 128 rows/block

    __shared__ float sW[K * N];
    for (int idx = threadIdx.x; idx < K * N; idx += 256) sW[idx] = W[idx];
    __syncthreads();

    const int wave = threadIdx.x >> 5;
    const int lane = threadIdx.x & 31;
    const int colTile = wave % COL_TILES;
    const int rowGroup = wave / COL_TILES;
    const int col0 = colTile * 16;
    const int lm = lane & 15;
    const int hi = lane >> 4;                     // lanes 16-31 hold K+2,K+3
    const int n = col0 + lm;

    // B fragments for this wave's 16 columns, cached in registers (reused per row tile)
    v2f bfrag[KSTEPS];
#pragma unroll
    for (int j = 0; j < KSTEPS; ++j) {
        const int ka = 4 * j + hi * 2;
        v2f b;
        b.x = sW[ka * N + n];
        b.y = sW[(ka + 1) * N + n];
        bfrag[j] = b;
    }
    const float bv = HASBIAS ? bias[n] : 0.0f;

    const int rowBase = blockIdx.x * ROWS + rowGroup * 16 * TILES_PER_WAVE;

#pragma unroll
    for (int rt = 0; rt < TILES_PER_WAVE; ++rt) {
        const int row0 = rowBase + rt * 16;
        const int m = row0 + lm;
        const int mc = (m < M) ? m : (M - 1);     // clamp: no divergence, store is masked
        const float* __restrict__ arow = A + (size_t)mc * K + hi * 2;

        v2f afrag[KSTEPS];
#pragma unroll
        for (int j = 0; j < KSTEPS; ++j) {        // 16 independent global_load_b64
            v2f a = *(const v2f*)(arow + 4 * j);
            if (RELU) { a.x = fmaxf(a.x, 0.0f); a.y = fmaxf(a.y, 0.0f); }
            afrag[j] = a;
        }

        v8f acc = {};
#pragma unroll
        for (int j = 0; j < KSTEPS; ++j) {        // dense WMMA chain on registers
            acc = __builtin_amdgcn_wmma_f32_16x16x4_f32(false, afrag[j], false, bfrag[j],
                                                        (short)0, acc, false, false);
        }

#pragma unroll
        for (int r = 0; r < 8; ++r) {
            const int mr = row0 + r + hi * 8;     // C: VGPR r -> M=r (lo lanes), M=r+8 (hi lanes)
            if (mr < M) C[(size_t)mr * N + n] = acc[r] + bv;
        }
    }
}

// ---------------- GCN hop: self-loop init + edge scatter ----------------

__global__ void hop_init_kernel(const float4* __restrict__ hw, const float* __restrict__ dinv,
                                const float* __restrict__ bh, float4* __restrict__ hnext, int M) {
    int t = blockIdx.x * blockDim.x + threadIdx.x;   // over M * (HID/4) float4s
    if (t >= M * (HID / 4)) return;
    int node = t >> 4;
    int c4 = (t & 15) * 4;
    float di = dinv[node];
    float d2 = di * di;
    float4 v = hw[t];
    float4 o;
    o.x = bh[c4 + 0] + v.x * d2;
    o.y = bh[c4 + 1] + v.y * d2;
    o.z = bh[c4 + 2] + v.z * d2;
    o.w = bh[c4 + 3] + v.w * d2;
    hnext[t] = o;                                    // self-loop message + bias
}

__global__ void scatter_kernel(const int* __restrict__ src, const int* __restrict__ dst,
                               const float* __restrict__ dinv, const float* __restrict__ hw,
                               float* __restrict__ hnext, int E) {
    int t = blockIdx.x * blockDim.x + threadIdx.x;
    int e = t >> 5;                                  // 32 lanes per edge, 2 channels/lane
    if (e >= E) return;
    int lane = t & 31;
    int s = src[e], d = dst[e];
    float w = dinv[s] * dinv[d];
    float2 v = *(const float2*)(hw + (size_t)s * HID + lane * 2);
    float* out = hnext + (size_t)d * HID + lane * 2;
    atomicAdd(out, v.x * w);
    atomicAdd(out + 1, v.y * w);
}

// ---------------- pooling ----------------

__global__ void zero_kernel(float* p, int n) {
    int i = blockIdx.x * blockDim.x + threadIdx.x;
    if (i < n) p[i] = 0.0f;
}

__global__ void pool_kernel(const float* __restrict__ logits, const int* __restrict__ batch,
                            float* sums, float* cnt, int M) {
    int t = blockIdx.x * blockDim.x + threadIdx.x;
    if (t >= M * OUTD) return;
    int node = t >> 5;
    int c = t & 31;
    int b = batch[node];
    atomicAdd(&sums[b * OUTD + c], logits[t]);
    if (c == 0) atomicAdd(&cnt[b], 1.0f);
}

__global__ void finalize_kernel(const float* __restrict__ sums, const float* __restrict__ cnt,
                                float* __restrict__ out) {
    int t = blockIdx.x * blockDim.x + threadIdx.x;
    if (t >= NGRAPHS * OUTD) return;
    out[t] = sums[t] / fmaxf(cnt[t >> 5], 1.0f);
}

// ---------------- driver ----------------

extern "C" void kernel_launch(void* const* d_in, const int* in_sizes, int n_in,
                              void* d_out, int out_size, void* d_ws, size_t ws_size,
                              hipStream_t stream) {
    (void)in_sizes; (void)n_in; (void)out_size; (void)ws_size;

    const float* x     = (const float*)d_in[0];
    const int*   eidx  = (const int*)d_in[1];   // (2, E): src row then dst row
    const int*   batch = (const int*)d_in[2];
    const float* W_in  = (const float*)d_in[3];
    const float* b_in  = (const float*)d_in[4];
    const float* W_h   = (const float*)d_in[5]; // (3, 64, 64)
    const float* b_h   = (const float*)d_in[6]; // (3, 64)
    const float* W_cls = (const float*)d_in[7];
    const float* b_cls = (const float*)d_in[8];

    const int* src = eidx;
    const int* dst = eidx + NEDGES;

    float* ws = (float*)d_ws;
    size_t off = 0;
    float* dinv = ws + off; off += 50048;                  // aligned 50000
    float* bufA = ws + off; off += (size_t)NNODES * HID;
    float* bufB = ws + off; off += (size_t)NNODES * HID;
    float* bufC = ws + off; off += (size_t)NNODES * HID;
    float* sums = ws + off; off += NGRAPHS * OUTD;
    float* cnt  = ws + off; off += NGRAPHS;

    const int T = 256;

    // symmetric normalization: dinv = rsqrt(1 + deg)
    deg_init_kernel<<<(NNODES + T - 1) / T, T, 0, stream>>>(dinv, NNODES);
    deg_edges_kernel<<<(NEDGES + T - 1) / T, T, 0, stream>>>(dst, dinv, NEDGES);
    dinv_kernel<<<(NNODES + T - 1) / T, T, 0, stream>>>(dinv, NNODES);

    // encoder: h0 = x @ W_in + b_in    (64 rows/block)
    gemm_bias_kernel<64, false, true><<<(NNODES + 63) / 64, T, 0, stream>>>(
        x, W_in, b_in, bufA, NNODES);

    float* h = bufA;
    float* hn = bufB;
    float* hw = bufC;
    for (int hop = 0; hop < NHOPS; ++hop) {
        // hw = relu?(h) @ W_h[hop]  (relu fused into A-load for hops 1,2)
        if (hop == 0) {
            gemm_bias_kernel<64, false, false><<<(NNODES + 63) / 64, T, 0, stream>>>(
                h, W_h + (size_t)hop * HID * HID, (const float*)nullptr, hw, NNODES);
        } else {
            gemm_bias_kernel<64, true, false><<<(NNODES + 63) / 64, T, 0, stream>>>(
                h, W_h + (size_t)hop * HID * HID, (const float*)nullptr, hw, NNODES);
        }
        // hn = b_h[hop] + hw * dinv^2 (self-loop), then edge scatter-add
        hop_init_kernel<<<((NNODES * (HID / 4)) + T - 1) / T, T, 0, stream>>>(
            (const float4*)hw, dinv, b_h + (size_t)hop * HID, (float4*)hn, NNODES);
        scatter_kernel<<<(int)(((size_t)NEDGES * 32 + T - 1) / T), T, 0, stream>>>(
            src, dst, dinv, hw, hn, NEDGES);
        float* tmp = h; h = hn; hn = tmp;
    }

    // classifier: logits = h @ W_cls + b_cls  (no relu after last hop; 128 rows/block)
    float* logits = bufC;  // hw buffer is free now
    gemm_bias_kernel<32, false, true><<<(NNODES + 127) / 128, T, 0, stream>>>(
        h, W_cls, b_cls, logits, NNODES);

    // global mean pool
    zero_kernel<<<(NGRAPHS * OUTD + NGRAPHS + T - 1) / T, T, 0, stream>>>(
        sums, NGRAPHS * OUTD + NGRAPHS);
    pool_kernel<<<((NNODES * OUTD) + T - 1) / T, T, 0, stream>>>(logits, batch, sums, cnt, NNODES);
    finalize_kernel<<<(NGRAPHS * OUTD + T - 1) / T, T, 0, stream>>>(sums, cnt, (float*)d_out);
}